// ComposeImgLoss_68143951118339
// MI455X (gfx1250) — compile-verified
//
#include <hip/hip_runtime.h>
#include <hip/hip_bf16.h>
#include <math.h>

typedef __attribute__((ext_vector_type(2))) float v2f;
typedef __attribute__((ext_vector_type(8))) float v8f;

#define N_ 8
#define ATTRI_ 6
#define SQE_ 14
#define CH_ 4
#define HW_ 16384
#define PRED_TOTAL (N_*ATTRI_*SQE_*CH_*HW_)   /* 44,040,192 floats */

/* workspace layout, offsets in floats */
#define P1_OFF   0                        /* 1024: pass-1 per-block min/max   */
#define MNMX_OFF 1024                     /* 3: mn, mx, 1/(mx-mn)             */
#define CNT_OFF  1028                     /* 1008 ints: c0/c1/c2 per (n,s,ch) */
#define TL_OFF   2036                     /* 336: type_list                   */
#define LAW_OFF  2432                     /* 393216: gen_L+gen_A+gen_W        */
#define WSUM_OFF (LAW_OFF+393216)         /* 393216: weighted.sum(axis=1)     */
#define P2_OFF   (WSUM_OFF+393216)        /* 2048: pass-2 per-block min/max   */
#define WMNMX_OFF (P2_OFF+2048)           /* 3: wmn, wmx, 1/(wmx-wmn)         */
#define P3_OFF   (WMNMX_OFF+8)            /* 512: loss partials               */

__device__ __forceinline__ float sig10(float v) {
    /* sigmoid((v-0.9)*10) */
    return 1.0f / (1.0f + __expf((0.9f - v) * 10.0f));
}

/* ---- K0: zero the count accumulators (atomics target) -------------------- */
__global__ void k0_zero(float* ws) {
    int* cnt = (int*)(ws + CNT_OFF);
    for (int i = threadIdx.x; i < N_ * SQE_ * 9; i += blockDim.x) cnt[i] = 0;
}

/* ---- K1: global min/max of Pred (float4, grid-stride) -------------------- */
__global__ void k1_minmax(const float* __restrict__ pred, float* __restrict__ ws) {
    const float4* p4 = (const float4*)pred;
    int tid = blockIdx.x * blockDim.x + threadIdx.x;
    int stride = gridDim.x * blockDim.x;
    float mn = INFINITY, mx = -INFINITY;
    for (int i = tid; i < PRED_TOTAL / 4; i += stride) {
        float4 v = p4[i];
        mn = fminf(mn, fminf(fminf(v.x, v.y), fminf(v.z, v.w)));
        mx = fmaxf(mx, fmaxf(fmaxf(v.x, v.y), fmaxf(v.z, v.w)));
    }
    __shared__ float smn[256], smx[256];
    smn[threadIdx.x] = mn; smx[threadIdx.x] = mx;
    __syncthreads();
    for (int off = 128; off > 0; off >>= 1) {
        if (threadIdx.x < off) {
            smn[threadIdx.x] = fminf(smn[threadIdx.x], smn[threadIdx.x + off]);
            smx[threadIdx.x] = fmaxf(smx[threadIdx.x], smx[threadIdx.x + off]);
        }
        __syncthreads();
    }
    if (threadIdx.x == 0) {
        ws[P1_OFF + 2 * blockIdx.x]     = smn[0];
        ws[P1_OFF + 2 * blockIdx.x + 1] = smx[0];
    }
}

/* ---- K2: finalize Pred min/max ------------------------------------------- */
__global__ void k2_fin_minmax(float* ws) {
    float mn = INFINITY, mx = -INFINITY;
    for (int i = threadIdx.x; i < 512; i += blockDim.x) {
        mn = fminf(mn, ws[P1_OFF + 2 * i]);
        mx = fmaxf(mx, ws[P1_OFF + 2 * i + 1]);
    }
    __shared__ float smn[256], smx[256];
    smn[threadIdx.x] = mn; smx[threadIdx.x] = mx;
    __syncthreads();
    for (int off = 128; off > 0; off >>= 1) {
        if (threadIdx.x < off) {
            smn[threadIdx.x] = fminf(smn[threadIdx.x], smn[threadIdx.x + off]);
            smx[threadIdx.x] = fmaxf(smx[threadIdx.x], smx[threadIdx.x + off]);
        }
        __syncthreads();
    }
    if (threadIdx.x == 0) {
        ws[MNMX_OFF]     = smn[0];
        ws[MNMX_OFF + 1] = smx[0];
        ws[MNMX_OFF + 2] = 1.0f / (smx[0] - smn[0]);
    }
}

/* ---- K3: color-threshold counts (attr 0) + fused LAW sum (attrs 2,3,5) --- */
__global__ void k3_counts_law(const float* __restrict__ pred, float* __restrict__ ws) {
    const float mn  = ws[MNMX_OFF];
    const float dv  = ws[MNMX_OFF + 1] - mn;   /* division path: match jnp */
    const float inv = ws[MNMX_OFF + 2];
    int*   cnt = (int*)(ws + CNT_OFF);
    float* law = ws + LAW_OFF;

    int n  = blockIdx.x >> 6;                       /* 64 blocks per batch  */
    int hw = ((blockIdx.x & 63) << 8) + threadIdx.x;
    int lane = threadIdx.x & 31;

    __shared__ int scnt[SQE_ * 9];
    for (int i = threadIdx.x; i < SQE_ * 9; i += blockDim.x) scnt[i] = 0;
    __syncthreads();

    float l0 = 0.f, l1 = 0.f, l2 = 0.f;
    for (int s = 0; s < SQE_; s++) {
        /* attr 0: counts */
        size_t b0 = ((((size_t)n * ATTRI_ + 0) * SQE_ + s) * CH_) * HW_ + hw;
        float r  = (pred[b0          ] - mn) / dv;
        float g  = (pred[b0 +   HW_  ] - mn) / dv;
        float b  = (pred[b0 + 2*HW_  ] - mn) / dv;
        float al = (pred[b0 + 3*HW_  ] - mn) / dv;
        bool a8 = al > 0.8f;
        float rgb[3] = { r, g, b };
        #pragma unroll
        for (int c = 0; c < 3; c++) {
            float v = rgb[c];
            unsigned long long m0 = __ballot(v > 0.0f && v < 0.2f && a8);
            unsigned long long m1 = __ballot(v > 0.4f && v < 0.6f);
            unsigned long long m2 = __ballot(v > 0.8f);
            if (lane == 0) {
                atomicAdd(&scnt[(s * 3 + c) * 3 + 0], (int)__popcll(m0));
                atomicAdd(&scnt[(s * 3 + c) * 3 + 1], (int)__popcll(m1));
                atomicAdd(&scnt[(s * 3 + c) * 3 + 2], (int)__popcll(m2));
            }
        }
        /* attrs 2,3,5 channels 0..2: fused L+A+W accumulation */
        #pragma unroll
        for (int ai = 0; ai < 3; ai++) {
            const int a = (ai == 0) ? 2 : ((ai == 1) ? 3 : 5);
            size_t ba = ((((size_t)n * ATTRI_ + a) * SQE_ + s) * CH_) * HW_ + hw;
            l0 += (pred[ba         ] - mn) * inv;
            l1 += (pred[ba +   HW_ ] - mn) * inv;
            l2 += (pred[ba + 2*HW_ ] - mn) * inv;
        }
    }
    law[((size_t)n * 3 + 0) * HW_ + hw] = l0;
    law[((size_t)n * 3 + 1) * HW_ + hw] = l1;
    law[((size_t)n * 3 + 2) * HW_ + hw] = l2;

    __syncthreads();
    for (int i = threadIdx.x; i < SQE_ * 9; i += blockDim.x)
        atomicAdd(&cnt[n * SQE_ * 9 + i], scnt[i]);
}

/* ---- K4: counts -> type_list (COLOR_LIST membership via code bitmask) ---- */
__global__ void k4_typelist(float* ws) {
    int t = threadIdx.x;
    if (t >= N_ * SQE_) return;
    int n = t / SQE_, s = t % SQE_;
    const int* c = (const int*)(ws + CNT_OFF) + (n * SQE_ + s) * 9;
    float vals[3];
    int code = 0;
    #pragma unroll
    for (int ch = 0; ch < 3; ch++) {
        int c0 = c[ch * 3 + 0], c1 = c[ch * 3 + 1], c2 = c[ch * 3 + 2];
        float v = (c2 > c1 && c2 > c0) ? 1.0f : ((c1 > c0) ? 0.5f : 0.0f);
        vals[ch] = v;
        code = code * 3 + ((v == 1.0f) ? 2 : ((v == 0.5f) ? 1 : 0));
    }
    /* base-3 codes of the 11 COLOR_LIST rows */
    const unsigned mask = (1u<<0)|(1u<<8)|(1u<<13)|(1u<<14)|(1u<<15)|(1u<<16)|
                          (1u<<19)|(1u<<20)|(1u<<22)|(1u<<24)|(1u<<26);
    bool member = (mask >> code) & 1u;
    float* tl = ws + TL_OFF;
    #pragma unroll
    for (int ch = 0; ch < 3; ch++)
        tl[(n * SQE_ + s) * 3 + ch] = member ? vals[ch] : 0.0f;
}

/* ---- K5: W(3xHW) = TL(3x14) x RS(14xHW) via V_WMMA_F32_16X16X4_F32 ------- */
/* One wave owns 16 pixels of one batch n. K accumulated in 4 chunks of 4.   */
__global__ void k5_weighted_wmma(const float* __restrict__ pred, float* __restrict__ ws) {
    const float mn  = ws[MNMX_OFF];
    const float inv = ws[MNMX_OFF + 2];
    const float* tl = ws + TL_OFF;
    float* wsum = ws + WSUM_OFF;

    int wid  = (blockIdx.x * blockDim.x + threadIdx.x) >> 5; /* global wave   */
    int lane = threadIdx.x & 31;
    int n    = wid >> 10;                                    /* 1024 tiles/n  */
    int tile = wid & 1023;
    int nn   = lane & 15;
    int half = lane >> 4;
    int p    = tile * 16 + nn;

    v8f acc = {0.f, 0.f, 0.f, 0.f, 0.f, 0.f, 0.f, 0.f};
    #pragma unroll
    for (int kc = 0; kc < 4; kc++) {
        int s0 = kc * 4 + 2 * half;   /* A/B slot 0 holds K = 2*half   */
        int s1 = s0 + 1;              /* slot 1 holds K = 2*half + 1   */
        float a0 = (nn < 3 && s0 < SQE_) ? tl[(n * SQE_ + s0) * 3 + nn] : 0.f;
        float a1 = (nn < 3 && s1 < SQE_) ? tl[(n * SQE_ + s1) * 3 + nn] : 0.f;
        float b0 = 0.f, b1 = 0.f;
        if (s0 < SQE_) {
            size_t ba = ((((size_t)n * ATTRI_ + 4) * SQE_ + s0) * CH_) * HW_ + p;
            float x0 = (pred[ba] - mn) * inv;
            float x1 = (pred[ba + HW_] - mn) * inv;
            float x2 = (pred[ba + 2 * HW_] - mn) * inv;
            b0 = sig10(x0) * sig10(x1) * sig10(x2);
        }
        if (s1 < SQE_) {
            size_t ba = ((((size_t)n * ATTRI_ + 4) * SQE_ + s1) * CH_) * HW_ + p;
            float x0 = (pred[ba] - mn) * inv;
            float x1 = (pred[ba + HW_] - mn) * inv;
            float x2 = (pred[ba + 2 * HW_] - mn) * inv;
            b1 = sig10(x0) * sig10(x1) * sig10(x2);
        }
        v2f A = {a0, a1};
        v2f B = {b0, b1};
        acc = __builtin_amdgcn_wmma_f32_16x16x4_f32(false, A, false, B,
                                                    (short)0, acc, false, false);
    }
    /* D rows 0..2 live in acc[0..2] of lanes 0..15 */
    if (half == 0) {
        wsum[((size_t)n * 3 + 0) * HW_ + p] = acc[0];
        wsum[((size_t)n * 3 + 1) * HW_ + p] = acc[1];
        wsum[((size_t)n * 3 + 2) * HW_ + p] = acc[2];
    }
    float lmn = (half == 0) ? fminf(acc[0], fminf(acc[1], acc[2])) :  INFINITY;
    float lmx = (half == 0) ? fmaxf(acc[0], fmaxf(acc[1], acc[2])) : -INFINITY;
    __shared__ float smn[256], smx[256];
    smn[threadIdx.x] = lmn; smx[threadIdx.x] = lmx;
    __syncthreads();
    for (int off = 128; off > 0; off >>= 1) {
        if (threadIdx.x < off) {
            smn[threadIdx.x] = fminf(smn[threadIdx.x], smn[threadIdx.x + off]);
            smx[threadIdx.x] = fmaxf(smx[threadIdx.x], smx[threadIdx.x + off]);
        }
        __syncthreads();
    }
    if (threadIdx.x == 0) {
        ws[P2_OFF + 2 * blockIdx.x]     = smn[0];
        ws[P2_OFF + 2 * blockIdx.x + 1] = smx[0];
    }
}

/* ---- K6: finalize weighted-sum min/max ----------------------------------- */
__global__ void k6_fin_wminmax(float* ws) {
    float mn = INFINITY, mx = -INFINITY;
    for (int i = threadIdx.x; i < 1024; i += blockDim.x) {
        mn = fminf(mn, ws[P2_OFF + 2 * i]);
        mx = fmaxf(mx, ws[P2_OFF + 2 * i + 1]);
    }
    __shared__ float smn[256], smx[256];
    smn[threadIdx.x] = mn; smx[threadIdx.x] = mx;
    __syncthreads();
    for (int off = 128; off > 0; off >>= 1) {
        if (threadIdx.x < off) {
            smn[threadIdx.x] = fminf(smn[threadIdx.x], smn[threadIdx.x + off]);
            smx[threadIdx.x] = fmaxf(smx[threadIdx.x], smx[threadIdx.x + off]);
        }
        __syncthreads();
    }
    if (threadIdx.x == 0) {
        ws[WMNMX_OFF]     = smn[0];
        ws[WMNMX_OFF + 1] = smx[0];
        ws[WMNMX_OFF + 2] = 1.0f / (smx[0] - smn[0]);
    }
}

/* ---- K7: composite + MSE partials ---------------------------------------- */
__global__ void k7_loss(const float* __restrict__ gt, float* __restrict__ ws) {
    const float wmn  = ws[WMNMX_OFF];
    const float winv = ws[WMNMX_OFF + 2];
    int n  = blockIdx.x >> 6;
    int hw = ((blockIdx.x & 63) << 8) + threadIdx.x;
    float acc = 0.f;
    #pragma unroll
    for (int c = 0; c < 3; c++) {
        float rc   = (ws[WSUM_OFF + ((size_t)n * 3 + c) * HW_ + hw] - wmn) * winv;
        float comp = rc + ws[LAW_OFF + ((size_t)n * 3 + c) * HW_ + hw];
        comp = fminf(fmaxf(comp, 0.0f), 1.0f);
        float gtn = (gt[((size_t)n * 4 + c) * HW_ + hw] + 1.0f) * 0.5f;
        float d = comp - gtn;
        acc += d * d;
    }
    __shared__ float ssum[256];
    ssum[threadIdx.x] = acc;
    __syncthreads();
    for (int off = 128; off > 0; off >>= 1) {
        if (threadIdx.x < off) ssum[threadIdx.x] += ssum[threadIdx.x + off];
        __syncthreads();
    }
    if (threadIdx.x == 0) ws[P3_OFF + blockIdx.x] = ssum[0];
}

/* ---- K8: final mean ------------------------------------------------------- */
__global__ void k8_fin_loss(const float* __restrict__ ws, float* __restrict__ out) {
    float acc = 0.f;
    for (int i = threadIdx.x; i < 512; i += blockDim.x) acc += ws[P3_OFF + i];
    __shared__ float ssum[256];
    ssum[threadIdx.x] = acc;
    __syncthreads();
    for (int off = 128; off > 0; off >>= 1) {
        if (threadIdx.x < off) ssum[threadIdx.x] += ssum[threadIdx.x + off];
        __syncthreads();
    }
    if (threadIdx.x == 0) out[0] = ssum[0] * (1.0f / (float)(N_ * 3 * HW_));
}

extern "C" void kernel_launch(void* const* d_in, const int* in_sizes, int n_in,
                              void* d_out, int out_size, void* d_ws, size_t ws_size,
                              hipStream_t stream) {
    const float* gt   = (const float*)d_in[0];   /* (8,4,128,128)           */
    const float* pred = (const float*)d_in[1];   /* (8,6,14,4,128,128)      */
    float* out = (float*)d_out;                  /* scalar loss             */
    float* ws  = (float*)d_ws;

    k0_zero        <<<   1, 256, 0, stream>>>(ws);
    k1_minmax      <<< 512, 256, 0, stream>>>(pred, ws);
    k2_fin_minmax  <<<   1, 256, 0, stream>>>(ws);
    k3_counts_law  <<< 512, 256, 0, stream>>>(pred, ws);
    k4_typelist    <<<   1, 128, 0, stream>>>(ws);
    k5_weighted_wmma<<<1024, 256, 0, stream>>>(pred, ws);
    k6_fin_wminmax <<<   1, 256, 0, stream>>>(ws);
    k7_loss        <<< 512, 256, 0, stream>>>(gt, ws);
    k8_fin_loss    <<<   1, 256, 0, stream>>>(ws, out);
}